// MultiplexGNN_20950850469923
// MI455X (gfx1250) — compile-verified
//
#include <hip/hip_runtime.h>
#include <stdint.h>

typedef float v2f __attribute__((ext_vector_type(2)));
typedef float v8f __attribute__((ext_vector_type(8)));
typedef int v4i_b __attribute__((vector_size(16)));   // GCC-style vector, matches builtin sig

#define AS1 __attribute__((address_space(1)))
#define AS3 __attribute__((address_space(3)))

// ---------------------------------------------------------------- utilities

__global__ void zero_f32_kernel(float* __restrict__ p, long long n) {
  long long i = (long long)blockIdx.x * blockDim.x + threadIdx.x;
  if (i < n) p[i] = 0.0f;
}

__global__ void deg_count_kernel(const int* __restrict__ dst,
                                 float* __restrict__ deg, int E) {
  int e = blockIdx.x * blockDim.x + threadIdx.x;
  if (e < E) atomicAdd(&deg[dst[e]], 1.0f);
}

__global__ void dinv_kernel(float* __restrict__ deg, int n) {
  int i = blockIdx.x * blockDim.x + threadIdx.x;
  if (i < n) deg[i] = rsqrtf(deg[i] + 1.0f);
}

// Staged-B LDS layout: sB[kstep][tile][lane][2] floats, so that the B fragment
// for (kstep, tile) is ONE aligned ds_load_b64 per lane (no register repacking).
// Mapping from W[kk][c] (row-major KxN):
//   kstep = kk>>2, rem = kk&3, half = rem>>1, y = rem&1
//   lane  = (c&15) | (half<<4),  dst = ((kstep*4 + tile)*32 + lane)*2 + y
__device__ __forceinline__ int staged_idx(int kk, int c) {
  int kstep = kk >> 2;
  int rem = kk & 3;
  int lane = (c & 15) | ((rem >> 1) << 4);
  return (((kstep << 2) | (c >> 4)) * 32 + lane) * 2 + (rem & 1);
}

// ------------------------------------------------- GEMM: h = A[n,64] @ W[64,64]
// One wave = 16 output rows x 64 cols (4 accumulators), fp32 WMMA 16x16x4.
__global__ __launch_bounds__(256) void gemm64_wmma_kernel(
    const float* __restrict__ A, const float* __restrict__ W,
    float* __restrict__ out, int n) {
  __shared__ __align__(16) float rawW[64 * 64];   // 16 KB, verbatim copy of W
  __shared__ __align__(16) float sB[64 * 64];     // 16 KB, fragment-native layout

#if __has_builtin(__builtin_amdgcn_global_load_async_to_lds_b128) && \
    __has_builtin(__builtin_amdgcn_s_wait_asynccnt)
  {
    // CDNA5 async copy: 4096 floats = 1024 x b128, 4 per thread (ASYNCcnt path).
    for (int i = 0; i < 4; ++i) {
      int t = threadIdx.x + i * 256;
      AS1 v4i_b* g = (AS1 v4i_b*)(uintptr_t)(W + t * 4);
      AS3 v4i_b* l = (AS3 v4i_b*)(uint32_t)(uintptr_t)(rawW + t * 4);
      __builtin_amdgcn_global_load_async_to_lds_b128(g, l, 0, 0);
    }
    __builtin_amdgcn_s_wait_asynccnt(0);
  }
#else
  for (int i = threadIdx.x; i < 64 * 64; i += 256) rawW[i] = W[i];
#endif
  __syncthreads();
  for (int idx = threadIdx.x; idx < 64 * 64; idx += 256)
    sB[staged_idx(idx >> 6, idx & 63)] = rawW[idx];
  __syncthreads();

  const int lane = threadIdx.x & 31;
  const int wave = threadIdx.x >> 5;
  const int rowTile = blockIdx.x * 8 + wave;
  if (rowTile * 16 >= n) return;  // wave-uniform: EXEC stays all-1s inside

  const int m = lane & 15;
  const int koff = (lane >> 4) << 1;
  const float* __restrict__ arow = A + (long long)(rowTile * 16 + m) * 64;
  const v2f* __restrict__ sBv = (const v2f*)sB;

  v8f acc0 = {}, acc1 = {}, acc2 = {}, acc3 = {};
#pragma unroll
  for (int kstep = 0; kstep < 16; ++kstep) {
    v2f a = *(const v2f*)(arow + (kstep << 2) + koff);
    const int base = (kstep << 7) + lane;  // (kstep*4 + tile)*32 + lane, tile=0
    v2f b0 = sBv[base];
    v2f b1 = sBv[base + 32];
    v2f b2 = sBv[base + 64];
    v2f b3 = sBv[base + 96];
    acc0 = __builtin_amdgcn_wmma_f32_16x16x4_f32(false, a, false, b0, (short)0, acc0, false, false);
    acc1 = __builtin_amdgcn_wmma_f32_16x16x4_f32(false, a, false, b1, (short)0, acc1, false, false);
    acc2 = __builtin_amdgcn_wmma_f32_16x16x4_f32(false, a, false, b2, (short)0, acc2, false, false);
    acc3 = __builtin_amdgcn_wmma_f32_16x16x4_f32(false, a, false, b3, (short)0, acc3, false, false);
  }

  const int crow = rowTile * 16 + ((lane >> 4) << 3);
  const int ccol = lane & 15;
  float* __restrict__ o = out + (long long)crow * 64 + ccol;
#pragma unroll
  for (int v = 0; v < 8; ++v) {
    o[(long long)v * 64 + 0]  = acc0[v];
    o[(long long)v * 64 + 16] = acc1[v];
    o[(long long)v * 64 + 32] = acc2[v];
    o[(long long)v * 64 + 48] = acc3[v];
  }
}

// ---------------------- final GEMM: out = concat(emb)[n,192] @ Wout[192,64] + b
__global__ __launch_bounds__(256) void gemmout_wmma_kernel(
    const float* __restrict__ emb, const float* __restrict__ Wout,
    const float* __restrict__ bout, float* __restrict__ out, int n) {
  __shared__ __align__(16) float sB[192 * 64];  // 48 KB fragment-native layout
  for (int idx = threadIdx.x; idx < 192 * 64; idx += 256)
    sB[staged_idx(idx >> 6, idx & 63)] = Wout[idx];
  __syncthreads();

  const int lane = threadIdx.x & 31;
  const int wave = threadIdx.x >> 5;
  const int rowTile = blockIdx.x * 8 + wave;
  if (rowTile * 16 >= n) return;

  const int m = lane & 15;
  const int koff = (lane >> 4) << 1;
  const long long chunk = (long long)n * 64;
  const float* __restrict__ arow0 = emb + (long long)(rowTile * 16 + m) * 64;
  const v2f* __restrict__ sBv = (const v2f*)sB;

  v8f acc0 = {}, acc1 = {}, acc2 = {}, acc3 = {};
#pragma unroll 4
  for (int kstep = 0; kstep < 48; ++kstep) {
    // A gathered across the G=3 concatenated embedding chunks
    const float* __restrict__ arow =
        arow0 + (long long)(kstep >> 4) * chunk + ((kstep & 15) << 2);
    v2f a = *(const v2f*)(arow + koff);
    const int base = (kstep << 7) + lane;
    v2f b0 = sBv[base];
    v2f b1 = sBv[base + 32];
    v2f b2 = sBv[base + 64];
    v2f b3 = sBv[base + 96];
    acc0 = __builtin_amdgcn_wmma_f32_16x16x4_f32(false, a, false, b0, (short)0, acc0, false, false);
    acc1 = __builtin_amdgcn_wmma_f32_16x16x4_f32(false, a, false, b1, (short)0, acc1, false, false);
    acc2 = __builtin_amdgcn_wmma_f32_16x16x4_f32(false, a, false, b2, (short)0, acc2, false, false);
    acc3 = __builtin_amdgcn_wmma_f32_16x16x4_f32(false, a, false, b3, (short)0, acc3, false, false);
  }

  const int crow = rowTile * 16 + ((lane >> 4) << 3);
  const int ccol = lane & 15;
  const float b0c = bout[ccol];
  const float b1c = bout[16 + ccol];
  const float b2c = bout[32 + ccol];
  const float b3c = bout[48 + ccol];
  float* __restrict__ o = out + (long long)crow * 64 + ccol;
#pragma unroll
  for (int v = 0; v < 8; ++v) {
    o[(long long)v * 64 + 0]  = acc0[v] + b0c;
    o[(long long)v * 64 + 16] = acc1[v] + b1c;
    o[(long long)v * 64 + 32] = acc2[v] + b2c;
    o[(long long)v * 64 + 48] = acc3[v] + b3c;
  }
}

// -------------------------------------------- edge scatter (L2-resident atomics)
// 64 threads per edge: each thread owns one feature. Rows are 256B contiguous,
// so each gather is two coalesced wave32 cache lines out of L2.
__global__ __launch_bounds__(256) void scatter_kernel(
    const float* __restrict__ hw, const int* __restrict__ src,
    const int* __restrict__ dst, const float* __restrict__ dinv,
    float* __restrict__ agg, int E) {
  int e = blockIdx.x * 4 + (threadIdx.x >> 6);
  if (e >= E) return;
  int d = threadIdx.x & 63;
  int s = src[e];
  int t = dst[e];
  if (d == 0) __builtin_prefetch(hw + (long long)s * 64, 0, 0);  // global_prefetch_b8
  float norm = dinv[s] * dinv[t];
  atomicAdd(agg + (long long)t * 64 + d, hw[(long long)s * 64 + d] * norm);
}

// ------------------------ out = agg + hw*dinv^2 + b, optional ReLU (in-place ok)
__global__ void finalize_kernel(const float* __restrict__ agg,
                                const float* __restrict__ hw,
                                const float* __restrict__ dinv,
                                const float* __restrict__ b,
                                float* __restrict__ out, int n, int relu) {
  long long i = (long long)blockIdx.x * blockDim.x + threadIdx.x;
  long long total = (long long)n * 64;
  if (i >= total) return;
  int node = (int)(i >> 6);
  int d = (int)(i & 63);
  float di = dinv[node];
  float v = agg[i] + hw[i] * (di * di) + b[d];
  out[i] = relu ? fmaxf(v, 0.0f) : v;
}

// ---------------------------------------------------------------- orchestration

extern "C" void kernel_launch(void* const* d_in, const int* in_sizes, int n_in,
                              void* d_out, int out_size, void* d_ws, size_t ws_size,
                              hipStream_t stream) {
  (void)n_in; (void)out_size; (void)ws_size;

  const float* x    = (const float*)d_in[0];
  const int*   edges= (const int*)d_in[1];
  const float* W0   = (const float*)d_in[2];
  const float* b0   = (const float*)d_in[3];
  const float* W1   = (const float*)d_in[4];
  const float* b1   = (const float*)d_in[5];
  const float* W2   = (const float*)d_in[6];
  const float* b2   = (const float*)d_in[7];
  const float* Wout = (const float*)d_in[8];
  const float* bout = (const float*)d_in[9];

  const int N = in_sizes[0] / 64;
  const int G = in_sizes[2] / (64 * 64);   // = 3
  const int E = in_sizes[1] / (2 * G);
  const long long N64 = (long long)N * 64;

  float* ws   = (float*)d_ws;
  float* emb  = ws;                           // G * N64  (concat embeddings)
  float* hw   = emb + (long long)G * N64;     // N64      (GEMM output)
  float* agg  = hw + N64;                     // N64      (scatter target / activation)
  float* dinv = agg + N64;                    // N        (deg -> rsqrt in place)

  const int rowTiles      = (N + 15) / 16;
  const int gemmBlocks    = (rowTiles + 7) / 8;
  const int blocksN64     = (int)((N64 + 255) / 256);
  const int blocksN       = (N + 255) / 256;
  const int blocksE       = (E + 255) / 256;
  const int scatterBlocks = (E + 3) / 4;

  const float* Wg[3] = {W0, W1, W2};
  const float* bg[3] = {b0, b1, b2};

  for (int g = 0; g < G; ++g) {
    const int* src = edges + (long long)g * 2 * E;
    const int* dst = src + E;

    zero_f32_kernel<<<blocksN, 256, 0, stream>>>(dinv, (long long)N);
    deg_count_kernel<<<blocksE, 256, 0, stream>>>(dst, dinv, E);
    dinv_kernel<<<blocksN, 256, 0, stream>>>(dinv, N);

    for (int l = 0; l < 3; ++l) {
      const float* Wl = Wg[l] + (long long)g * 64 * 64;
      const float* bl = bg[l] + (long long)g * 64;
      const float* Ain = (l == 0) ? x : agg;  // agg holds previous activation

      gemm64_wmma_kernel<<<gemmBlocks, 256, 0, stream>>>(Ain, Wl, hw, N);
      zero_f32_kernel<<<blocksN64, 256, 0, stream>>>(agg, N64);
      scatter_kernel<<<scatterBlocks, 256, 0, stream>>>(hw, src, dst, dinv, agg, E);

      float* outp = (l == 2) ? (emb + (long long)g * N64) : agg;
      finalize_kernel<<<blocksN64, 256, 0, stream>>>(agg, hw, dinv, bl, outp, N,
                                                     (l < 2) ? 1 : 0);
    }
  }

  gemmout_wmma_kernel<<<gemmBlocks, 256, 0, stream>>>(emb, Wout, bout,
                                                      (float*)d_out, N);
}